// DifferenceOfGaussians_46067819217418
// MI455X (gfx1250) — compile-verified
//
#include <hip/hip_runtime.h>

typedef float v2f __attribute__((ext_vector_type(2)));
typedef float v8f __attribute__((ext_vector_type(8)));

#define IMG   512
#define PLANE (IMG*IMG)        // 262144
#define NSIG  14
#define NDOG  13
#define KSZ   87
#define RAD   43
#define THRESH 0.001f
#define MAX_PEAKS 32768

// padded 1D-tap LDS vector: index d+WPAD, d in [-15, 103] -> [0, 118]
#define WPAD  15
#define WLEN  120

// shared conv tile: 64 outputs + 86 halo = 150 valid, +2 zero pad
#define TVAL  150
#define TW    152

// workspace layout (in floats)
#define OFF_W1   0                       // 14*87 1D taps (region padded to 2048)
#define OFF_H    2048                    // 14 planes: horizontal pass (reused as DoG)
#define OFF_G    (OFF_H + NSIG*PLANE)    // 14 planes: full Gaussian pyramid
#define OFF_META (OFF_G + NSIG*PLANE)    // int region: counts[832], offsets[832], total

#define NBLK 832
#define ELEMS_PER_BLOCK 4096             // 832*4096 == 13*262144 exactly
#define ELEMS_PER_THREAD 16

// -------- K0: recover normalized 1D taps from the 2D outer-product kernels --------
// k2 = outer(w1,w1), w1 = g1/sum(g1); center row: k2[43,k] = w1_ctr*w1[k],
// k2[43,43] = w1_ctr^2  =>  w1[k] = k2[43,k]/sqrt(k2[43,43]).
__global__ void dog_wextract(const float* __restrict__ W, float* __restrict__ w1) {
    for (int i = threadIdx.x; i < NSIG * KSZ; i += blockDim.x) {
        int s = i / KSZ, t = i - s * KSZ;
        const float* base = W + s * KSZ * KSZ + RAD * KSZ;
        w1[i] = base[t] / sqrtf(base[RAD]);
    }
}

// -------- K1: horizontal 1D conv via V_WMMA_F32_16X16X4_F32 --------
// 4 waves/WG, each owns one 16x16 tile; shared 16x150 input strip + padded taps.
// O[m][c] = sum_u I[yb+m][x0+u] * wpad[u - c], band in B (zero outside via LDS pad).
__global__ __launch_bounds__(128) void dog_hconv(const float* __restrict__ x,
                                                 const float* __restrict__ w1,
                                                 float* __restrict__ h) {
    const int xb0 = blockIdx.x * 64, yb = blockIdx.y * 16, s = blockIdx.z;
    const int tid = threadIdx.x, lane = tid & 31, wv = tid >> 5;
    __shared__ float sWp[WLEN];
    __shared__ float sT[16][TW];

    for (int k = tid; k < WLEN; k += 128)
        sWp[k] = (k >= WPAD && k < WPAD + KSZ) ? w1[s * KSZ + (k - WPAD)] : 0.f;
    for (int k = tid; k < 16 * TW; k += 128) {
        int r = k / TW, a = k - r * TW;
        int gx = xb0 + a - RAD;
        float v = 0.f;
        if (a < TVAL && gx >= 0 && gx < IMG) v = x[(yb + r) * IMG + gx];
        sT[r][a] = v;                     // cols 150,151 zeroed
    }
    __syncthreads();

    const int m  = lane & 15;
    const int j0 = (lane < 16) ? 0 : 2;   // K-slot pair for this half-wave
    const int xoff = wv * 16;             // this wave's tile inside the strip
    v8f acc = {0.f, 0.f, 0.f, 0.f, 0.f, 0.f, 0.f, 0.f};
    for (int b = 0; b < 26; ++b) {
        int u0 = 4 * b + j0;
        v2f A, B;
        A.x = sT[m][xoff + u0];
        A.y = sT[m][xoff + u0 + 1];
        B.x = sWp[u0 - m + WPAD];         // branchless band tap (0 outside)
        B.y = sWp[u0 - m + WPAD + 1];
        acc = __builtin_amdgcn_wmma_f32_16x16x4_f32(false, A, false, B,
                                                    (short)0, acc, false, false);
    }
    float* out = h + s * PLANE;
    const int rbase = yb + ((lane >= 16) ? 8 : 0);
#pragma unroll
    for (int v = 0; v < 8; ++v)
        out[(rbase + v) * IMG + xb0 + xoff + m] = acc[v];
}

// -------- K2: vertical 1D conv via WMMA (band in A) --------
// 4 waves/WG stacked in y; shared 150x16 strip. O[m][c] = sum_u wpad[u-m]*H[y0+u-43][xb+c]
__global__ __launch_bounds__(128) void dog_vconv(const float* __restrict__ h,
                                                 const float* __restrict__ w1,
                                                 float* __restrict__ g) {
    const int xb = blockIdx.x * 16, yb0 = blockIdx.y * 64, s = blockIdx.z;
    const int tid = threadIdx.x, lane = tid & 31, wv = tid >> 5;
    __shared__ float sWp[WLEN];
    __shared__ float sH[TW][16];

    for (int k = tid; k < WLEN; k += 128)
        sWp[k] = (k >= WPAD && k < WPAD + KSZ) ? w1[s * KSZ + (k - WPAD)] : 0.f;
    const float* hp = h + s * PLANE;
    for (int k = tid; k < TW * 16; k += 128) {
        int u = k >> 4, c = k & 15;
        int gy = yb0 + u - RAD;
        float v = 0.f;
        if (u < TVAL && gy >= 0 && gy < IMG) v = hp[gy * IMG + xb + c];
        sH[u][c] = v;                     // rows 150,151 zeroed
    }
    __syncthreads();

    const int m  = lane & 15;
    const int j0 = (lane < 16) ? 0 : 2;
    const int yoff = wv * 16;
    v8f acc = {0.f, 0.f, 0.f, 0.f, 0.f, 0.f, 0.f, 0.f};
    for (int b = 0; b < 26; ++b) {
        int u0 = 4 * b + j0;
        v2f A, B;
        A.x = sWp[u0 - m + WPAD];
        A.y = sWp[u0 - m + WPAD + 1];
        B.x = sH[yoff + u0][m];
        B.y = sH[yoff + u0 + 1][m];
        acc = __builtin_amdgcn_wmma_f32_16x16x4_f32(false, A, false, B,
                                                    (short)0, acc, false, false);
    }
    float* out = g + s * PLANE;
    const int rbase = yb0 + yoff + ((lane >= 16) ? 8 : 0);
#pragma unroll
    for (int v = 0; v < 8; ++v)
        out[(rbase + v) * IMG + xb + m] = acc[v];
}

// -------- K3: DoG = (g[s] - g[s+1]) * sigma[s] --------
__global__ void dog_diff(const float* __restrict__ g, const float* __restrict__ sig,
                         float* __restrict__ dog) {
    int idx = blockIdx.x * blockDim.x + threadIdx.x;
    if (idx >= NDOG * PLANE) return;
    int s = idx / PLANE;
    dog[idx] = (g[idx] - g[idx + PLANE]) * sig[s];
}

// -------- peak predicate: interior (border-excluded) 3x3x3 max + threshold --------
__device__ __forceinline__ bool is_peak(const float* __restrict__ dog, int idx) {
    int s = idx / PLANE;
    int rem = idx - s * PLANE;
    int y = rem >> 9, xx = rem & 511;
    if (s < 1 || s > NDOG - 2 || y < 1 || y > IMG - 2 || xx < 1 || xx > IMG - 2)
        return false;                     // after exclusion all 26 neighbors in-bounds
    float d = dog[idx];
    if (!(d > THRESH)) return false;
#pragma unroll
    for (int ds = -1; ds <= 1; ++ds)
#pragma unroll
        for (int dy = -1; dy <= 1; ++dy)
#pragma unroll
            for (int dx = -1; dx <= 1; ++dx) {
                if (ds == 0 && dy == 0 && dx == 0) continue;
                if (dog[idx + ds * PLANE + dy * IMG + dx] > d) return false;
            }
    return true;
}

// -------- K4: per-block peak counts --------
__global__ __launch_bounds__(256) void dog_count(const float* __restrict__ dog,
                                                 int* __restrict__ meta) {
    int tid = threadIdx.x;
    int base = blockIdx.x * ELEMS_PER_BLOCK + tid * ELEMS_PER_THREAD;
    int c = 0;
    for (int e = 0; e < ELEMS_PER_THREAD; ++e)
        if (is_peak(dog, base + e)) ++c;
    __shared__ int sc;
    if (tid == 0) sc = 0;
    __syncthreads();
    if (c) atomicAdd(&sc, c);
    __syncthreads();
    if (tid == 0) meta[blockIdx.x] = sc;
}

// -------- K5: tiny serial exclusive scan (832 ints) --------
__global__ void dog_scan(int* __restrict__ meta) {
    int run = 0;
    for (int b = 0; b < NBLK; ++b) { meta[NBLK + b] = run; run += meta[b]; }
    meta[2 * NBLK] = run;
}

// -------- K6: ordered scatter of [sigma, y, x] rows --------
__global__ __launch_bounds__(256) void dog_scatter(const float* __restrict__ dog,
                                                   const int* __restrict__ meta,
                                                   const float* __restrict__ sig,
                                                   float* __restrict__ out) {
    int tid = threadIdx.x;
    int base = blockIdx.x * ELEMS_PER_BLOCK + tid * ELEMS_PER_THREAD;
    unsigned pm = 0;                      // peak bitmask for this thread's 16 elems
    for (int e = 0; e < ELEMS_PER_THREAD; ++e)
        if (is_peak(dog, base + e)) pm |= (1u << e);
    int c = __popc(pm);
    __shared__ int sc[256];
    sc[tid] = c;
    __syncthreads();
    for (int off = 1; off < 256; off <<= 1) {   // Hillis-Steele inclusive scan
        int v = (tid >= off) ? sc[tid - off] : 0;
        __syncthreads();
        sc[tid] += v;
        __syncthreads();
    }
    int pos = meta[NBLK + blockIdx.x] + (sc[tid] - c);  // exclusive prefix
    for (int e = 0; e < ELEMS_PER_THREAD; ++e) {
        if (pm & (1u << e)) {
            if (pos < MAX_PEAKS) {
                int idx = base + e;
                int s = idx / PLANE;
                int rem = idx - s * PLANE;
                out[pos * 3 + 0] = sig[s];
                out[pos * 3 + 1] = (float)(rem >> 9);
                out[pos * 3 + 2] = (float)(rem & 511);
            }
            ++pos;
        }
    }
}

// -------- K7: pad unused rows with [sigma[0], 0, 0] (nonzero fill_value = 0) --------
__global__ void dog_fill(const int* __restrict__ meta, const float* __restrict__ sig,
                         float* __restrict__ out) {
    int j = blockIdx.x * blockDim.x + threadIdx.x;
    if (j >= MAX_PEAKS) return;
    int total = meta[2 * NBLK];
    if (total > MAX_PEAKS) total = MAX_PEAKS;
    if (j >= total) {
        out[j * 3 + 0] = sig[0];
        out[j * 3 + 1] = 0.f;
        out[j * 3 + 2] = 0.f;
    }
}

extern "C" void kernel_launch(void* const* d_in, const int* in_sizes, int n_in,
                              void* d_out, int out_size, void* d_ws, size_t ws_size,
                              hipStream_t stream) {
    (void)in_sizes; (void)n_in; (void)out_size; (void)ws_size;
    const float* x   = (const float*)d_in[0];   // [1,1,512,512]
    const float* W   = (const float*)d_in[1];   // [14,1,87,87]
    const float* sig = (const float*)d_in[2];   // [14]
    float* out = (float*)d_out;                 // [32768,3]
    float* ws  = (float*)d_ws;

    float* w1   = ws + OFF_W1;
    float* hbuf = ws + OFF_H;
    float* gbuf = ws + OFF_G;
    float* dogb = hbuf;                         // reuse after vconv
    int*   meta = (int*)(ws + OFF_META);

    dog_wextract<<<1, 256, 0, stream>>>(W, w1);
    dog_hconv<<<dim3(8, 32, NSIG), 128, 0, stream>>>(x, w1, hbuf);
    dog_vconv<<<dim3(32, 8, NSIG), 128, 0, stream>>>(hbuf, w1, gbuf);
    dog_diff<<<(NDOG * PLANE + 255) / 256, 256, 0, stream>>>(gbuf, sig, dogb);
    dog_count<<<NBLK, 256, 0, stream>>>(dogb, meta);
    dog_scan<<<1, 1, 0, stream>>>(meta);
    dog_scatter<<<NBLK, 256, 0, stream>>>(dogb, meta, sig, out);
    dog_fill<<<(MAX_PEAKS + 255) / 256, 256, 0, stream>>>(meta, sig, out);
}